// ACEEvaluator_87436944212793
// MI455X (gfx1250) — compile-verified
//
#include <hip/hip_runtime.h>
#include <hip/hip_bf16.h>

#define N_ATOMS 1024
#define N_SPECIES 4
#define F_RS 64
#define D_SPH 1024
#define M2 20000
#define M3 40000
#define M4 60000
#define BLOCK 256

typedef unsigned int v4u __attribute__((ext_vector_type(4)));
typedef int          v8i __attribute__((ext_vector_type(8)));
typedef int          v4i __attribute__((ext_vector_type(4)));
typedef float        v2f __attribute__((ext_vector_type(2)));
typedef float        v8f __attribute__((ext_vector_type(8)));

#if defined(__has_builtin)
#  if __has_builtin(__builtin_amdgcn_tensor_load_to_lds) && __has_builtin(__builtin_amdgcn_s_wait_tensorcnt)
#    define HAVE_TDM 1
#  endif
#  if __has_builtin(__builtin_amdgcn_wmma_f32_16x16x4_f32)
#    define HAVE_WMMA 1
#  endif
#endif

// ---------------------------------------------------------------------------
// Kernel 1: composition energies.  e = (sum of 4 comp values) * mult0, summed
// over all atoms into *out via per-wave atomics.
// ---------------------------------------------------------------------------
__global__ __launch_bounds__(256) void ace_comp_kernel(
    const float* __restrict__ comp, const float* __restrict__ mult0,
    float* __restrict__ out)
{
    const int a = blockIdx.x * blockDim.x + threadIdx.x;
    float e = 0.f;
    if (a < N_ATOMS) {
        float4 c = reinterpret_cast<const float4*>(comp)[a];   // 4 contiguous per atom
        e = (c.x + c.y + c.z + c.w) * mult0[0];
    }
    // wave32 reduction
    for (int off = 16; off > 0; off >>= 1) e += __shfl_down(e, off, 32);
    if ((threadIdx.x & 31) == 0) atomicAdd(out, e);
}

// ---------------------------------------------------------------------------
// Kernel 2: radial energies via WMMA.
//   E_r = sum_{a,f} radial[a,f] * mult1[spec[a], f]
// Per 16-atom block: A(16x4) = one-hot(species), B(4x16) = mult1 chunk,
// D = A x B  ==>  D[m,n] = mult1[spec[atom_m], f_n]  (species gather done by
// the matrix pipe), then lane-wise FMA with radial in the same C/D layout.
// One wave (32 lanes) per block; 64 blocks cover 1024 atoms.
// ---------------------------------------------------------------------------
__global__ __launch_bounds__(32) void ace_radial_wmma_kernel(
    const float* __restrict__ radial, const float* __restrict__ mult1,
    const int* __restrict__ rad_species, float* __restrict__ out)
{
    const int ab   = blockIdx.x;        // 16-atom block index
    const int lane = threadIdx.x;       // 0..31
    const int mrow = lane & 15;
    const bool hi  = lane >= 16;
    float acc = 0.f;

#if defined(HAVE_WMMA)
    // A 16x4 f32 layout: lanes 0-15 hold K={0,1}, lanes 16-31 hold K={2,3}, M=lane%16
    const int s  = rad_species[ab * 16 + mrow];
    const int k0 = hi ? 2 : 0;
    v2f a;
    a.x = (s == k0    ) ? 1.f : 0.f;
    a.y = (s == k0 + 1) ? 1.f : 0.f;

    for (int fc = 0; fc < 4; ++fc) {                 // 4 chunks of 16 features
        // B 4x16 f32: rows striped across lanes within a VGPR
        const int kb0 = hi ? 1 : 0;
        const int kb1 = hi ? 3 : 2;
        v2f b;
        b.x = mult1[kb0 * F_RS + fc * 16 + mrow];
        b.y = mult1[kb1 * F_RS + fc * 16 + mrow];
        v8f c = {};
        c = __builtin_amdgcn_wmma_f32_16x16x4_f32(
                /*neg_a=*/false, a, /*neg_b=*/false, b,
                /*c_mod=*/(short)0, c, /*reuse_a=*/false, /*reuse_b=*/false);
        // C/D layout: VGPR v, lanes 0-15 -> M=v, lanes 16-31 -> M=v+8; N=lane%16
        #pragma unroll
        for (int v = 0; v < 8; ++v) {
            const int am = ab * 16 + v + (hi ? 8 : 0);
            acc = fmaf(c[v], radial[am * F_RS + fc * 16 + mrow], acc);
        }
    }
#else
    // scalar fallback: 16 atoms x 64 features over 32 lanes
    for (int idx = lane; idx < 16 * F_RS; idx += 32) {
        const int m = idx >> 6, f = idx & 63;
        const int am = ab * 16 + m;
        acc = fmaf(radial[am * F_RS + f], mult1[rad_species[am] * F_RS + f], acc);
    }
#endif

    for (int off = 16; off > 0; off >>= 1) acc += __shfl_down(acc, off, 32);
    if (lane == 0) atomicAdd(out, acc);
}

// ---------------------------------------------------------------------------
// Kernel 3: polynomial terms e2+e3+e4.  One workgroup per atom.  The 4KB sph
// row is DMA'd into LDS by the Tensor Data Mover (wave 0 issues, waits on
// TENSORcnt), then 256 threads stream coalesced index/mult arrays from L2 and
// gather products out of LDS.
// ---------------------------------------------------------------------------
__global__ __launch_bounds__(256) void ace_poly_kernel(
    const float* __restrict__ sph,
    const int* __restrict__ c2, const float* __restrict__ m2,
    const int* __restrict__ c3, const float* __restrict__ m3,
    const int* __restrict__ c4, const float* __restrict__ m4,
    const int* __restrict__ sph_species, float* __restrict__ out)
{
    __shared__ float s_sph[D_SPH];
    __shared__ float s_red[8];

    const int atom = blockIdx.x;
    const float* __restrict__ row = sph + (size_t)atom * D_SPH;

#if defined(HAVE_TDM)
    if (threadIdx.x < 32) {                     // wave 0 issues the TDM descriptor
        const unsigned lds_off = (unsigned)(size_t)&s_sph[0];
        const unsigned long long ga = (unsigned long long)(size_t)row;
        // D# group 0: count=1 | lds_addr | global_addr[56:0] | type=2
        v4u g0;
        g0.x = 1u;
        g0.y = lds_off;
        g0.z = (unsigned)(ga & 0xFFFFFFFFull);
        g0.w = (unsigned)((ga >> 32) & 0x01FFFFFFull) | (2u << 30);
        // D# group 1: data_size=4B, tensor_dim0=1024, tensor_dim1=1,
        //             tile_dim0=1024, tile_dim1=1, dim0_stride=1024
        v8i g1;
        g1[0] = (int)(2u << 16);                          // data_size = 2 (4 bytes)
        g1[1] = (int)(((unsigned)D_SPH & 0xFFFFu) << 16); // tensor_dim0 lo16
        g1[2] = (int)(1u << 16);                          // tensor_dim0 hi=0 | tensor_dim1=1
        g1[3] = (int)(((unsigned)D_SPH) << 16);           // tile_dim0 = 1024
        g1[4] = 1;                                        // tile_dim1 = 1
        g1[5] = D_SPH;                                    // tensor_dim0_stride lo32
        g1[6] = 0;
        g1[7] = 0;
        const v4i gz4 = {0, 0, 0, 0};                     // groups 2/3 unused (<=2D)
        const v8i gz8 = {0, 0, 0, 0, 0, 0, 0, 0};         // extra group (clang-23 6-arg form)
        __builtin_amdgcn_tensor_load_to_lds(g0, g1, gz4, gz4, gz8, 0);
        __builtin_amdgcn_s_wait_tensorcnt(0);
    }
    __syncthreads();
#else
    // cooperative float4 staging: 256 threads x 16B = 4KB
    reinterpret_cast<float4*>(s_sph)[threadIdx.x] =
        reinterpret_cast<const float4*>(row)[threadIdx.x];
    __syncthreads();
#endif

    const int sp = sph_species[atom];
    float acc = 0.f;

    {   // e2: 2-way products
        const float* __restrict__ mr = m2 + (size_t)sp * M2;
        #pragma unroll 2
        for (int m = threadIdx.x; m < M2; m += BLOCK) {
            const float p = s_sph[c2[m]] * s_sph[c2[M2 + m]];
            acc = fmaf(p, mr[m], acc);
        }
    }
    {   // e3: 3-way products
        const float* __restrict__ mr = m3 + (size_t)sp * M3;
        #pragma unroll 2
        for (int m = threadIdx.x; m < M3; m += BLOCK) {
            const float p = s_sph[c3[m]] * s_sph[c3[M3 + m]] * s_sph[c3[2 * M3 + m]];
            acc = fmaf(p, mr[m], acc);
        }
    }
    {   // e4: 4-way products
        const float* __restrict__ mr = m4 + (size_t)sp * M4;
        #pragma unroll 2
        for (int m = threadIdx.x; m < M4; m += BLOCK) {
            const float p = (s_sph[c4[m]] * s_sph[c4[M4 + m]]) *
                            (s_sph[c4[2 * M4 + m]] * s_sph[c4[3 * M4 + m]]);
            acc = fmaf(p, mr[m], acc);
        }
    }

    // wave32 reduce then cross-wave reduce in LDS
    for (int off = 16; off > 0; off >>= 1) acc += __shfl_down(acc, off, 32);
    if ((threadIdx.x & 31) == 0) s_red[threadIdx.x >> 5] = acc;
    __syncthreads();
    if (threadIdx.x == 0) {
        float t = 0.f;
        #pragma unroll
        for (int i = 0; i < 8; ++i) t += s_red[i];
        atomicAdd(out, t);
    }
}

// ---------------------------------------------------------------------------
extern "C" void kernel_launch(void* const* d_in, const int* in_sizes, int n_in,
                              void* d_out, int out_size, void* d_ws, size_t ws_size,
                              hipStream_t stream) {
    const float* comp        = (const float*)d_in[0];
    const float* radial      = (const float*)d_in[1];
    const float* sph         = (const float*)d_in[2];
    const float* mult0       = (const float*)d_in[3];
    const float* mult1       = (const float*)d_in[4];
    const int*   c2          = (const int*)  d_in[5];
    const float* m2          = (const float*)d_in[6];
    const int*   c3          = (const int*)  d_in[7];
    const float* m3          = (const float*)d_in[8];
    const int*   c4          = (const int*)  d_in[9];
    const float* m4          = (const float*)d_in[10];
    const int*   rad_species = (const int*)  d_in[11];
    const int*   sph_species = (const int*)  d_in[12];
    float* out = (float*)d_out;

    (void)hipMemsetAsync(out, 0, sizeof(float), stream);   // capture-safe memset node

    ace_comp_kernel<<<(N_ATOMS + 255) / 256, 256, 0, stream>>>(comp, mult0, out);
    ace_radial_wmma_kernel<<<N_ATOMS / 16, 32, 0, stream>>>(radial, mult1, rad_species, out);
    ace_poly_kernel<<<N_ATOMS, BLOCK, 0, stream>>>(sph, c2, m2, c3, m3, c4, m4,
                                                   sph_species, out);
    (void)in_sizes; (void)n_in; (void)out_size; (void)d_ws; (void)ws_size;
}